// Model_70549132804304
// MI455X (gfx1250) — compile-verified
//
#include <hip/hip_runtime.h>

typedef float f32x4 __attribute__((ext_vector_type(4)));  // native vector for NT stores

// ---------------------------------------------------------------------------
// Phase 1: inclusive scan of repeat counts (N = 65536, int32) into d_ws.
// Single 1024-thread block, Hillis-Steele in LDS, 64 chunks. Cost is noise
// next to the 512 MB output copy.
// ---------------------------------------------------------------------------
#define SCAN_BLOCK 1024

__global__ __launch_bounds__(SCAN_BLOCK) void scan_kernel(
    const int* __restrict__ r, int* __restrict__ cum, int n) {
  __shared__ int buf[SCAN_BLOCK];
  __shared__ int carry;
  const int tid = threadIdx.x;
  if (tid == 0) carry = 0;
  __syncthreads();
  for (int base = 0; base < n; base += SCAN_BLOCK) {
    int v = (base + tid < n) ? r[base + tid] : 0;
    buf[tid] = v;
    __syncthreads();
    for (int off = 1; off < SCAN_BLOCK; off <<= 1) {
      int self  = buf[tid];
      int other = (tid >= off) ? buf[tid - off] : 0;
      __syncthreads();
      buf[tid] = self + other;
      __syncthreads();
    }
    int c = carry;
    if (base + tid < n) cum[base + tid] = buf[tid] + c;
    __syncthreads();
    if (tid == SCAN_BLOCK - 1) carry = c + buf[SCAN_BLOCK - 1];
    __syncthreads();
  }
}

// ---------------------------------------------------------------------------
// Phase 2: gather. One block = ROWS_PER_BLOCK output rows. Binary search for
// the first row, linear advance for the rest. Row copy goes through the
// CDNA5 async copy engine: global -> LDS -> global, 128b per lane, tracked
// with ASYNCcnt. Output stores are non-temporal (write-once 512 MB stream >
// 192 MB L2) so L2 stays dedicated to cum[] and the re-read source rows.
// Tail rows (>= total) are zero-filled with non-temporal b128 stores.
// ---------------------------------------------------------------------------
#define GATHER_THREADS 128
#define ROWS_PER_BLOCK 4
#define MAX_D 512  // floats per row staged in LDS per row slot (2 KB)

__global__ __launch_bounds__(GATHER_THREADS) void gather_rows_kernel(
    const float* __restrict__ feeds, const int* __restrict__ cum,
    float* __restrict__ out, int n_src, int d, int out_rows) {
  __shared__ float smem[ROWS_PER_BLOCK * MAX_D];
  const int tid = threadIdx.x;
  const int row0 = blockIdx.x * ROWS_PER_BLOCK;
  const int total = cum[n_src - 1];
  const int vec_per_row = d >> 2;  // float4 chunks per row (512 -> 128)
  const bool use_async = (d <= MAX_D) && ((d & 3) == 0);

  // upper_bound(cum, row0): first src with cum[src] > row0
  int src = 0;
  if (row0 < total) {
    int lo = 0, hi = n_src;
    while (lo < hi) {
      int mid = (lo + hi) >> 1;
      if (cum[mid] <= row0) lo = mid + 1; else hi = mid;
    }
    src = lo;
  }

  int issued[ROWS_PER_BLOCK];
  int n_async = 0;
  for (int k = 0; k < ROWS_PER_BLOCK; ++k) {
    issued[k] = -1;
    const int j = row0 + k;
    if (j >= out_rows) break;
    if (j >= total) {
      // zero pad row; NT store keeps the streaming output out of L2
      f32x4 z = {0.f, 0.f, 0.f, 0.f};
      f32x4* drow = reinterpret_cast<f32x4*>(out + (size_t)j * d);
      for (int t = tid; t < vec_per_row; t += GATHER_THREADS)
        __builtin_nontemporal_store(z, &drow[t]);
      continue;
    }
    while (src < n_src - 1 && cum[src] <= j) ++src;  // monotone advance (+clip)
    const float* grow = feeds + (size_t)src * d;
    if (use_async) {
      for (int t = tid; t < vec_per_row; t += GATHER_THREADS) {
        // LDS aperture keeps the byte offset in addr[31:0]
        unsigned lds_addr = (unsigned)(size_t)(&smem[k * MAX_D + 4 * t]);
        unsigned long long gaddr =
            (unsigned long long)(size_t)(const void*)(grow + 4 * t);
        asm volatile("global_load_async_to_lds_b128 %0, %1, off"
                     :: "v"(lds_addr), "v"(gaddr) : "memory");
      }
      issued[k] = src;
      ++n_async;
    } else {
      // generic fallback (not taken for d == 512)
      float* drow = out + (size_t)j * d;
      for (int t = tid; t < d; t += GATHER_THREADS) drow[t] = grow[t];
    }
  }

  if (n_async) {
    // all per-wave async loads complete -> staged rows are in LDS
    asm volatile("s_wait_asynccnt 0" ::: "memory");
    for (int k = 0; k < ROWS_PER_BLOCK; ++k) {
      const int j = row0 + k;
      if (j >= out_rows) break;
      if (issued[k] < 0) continue;
      float* drow = out + (size_t)j * d;
      for (int t = tid; t < vec_per_row; t += GATHER_THREADS) {
        unsigned lds_addr = (unsigned)(size_t)(&smem[k * MAX_D + 4 * t]);
        unsigned long long gaddr =
            (unsigned long long)(size_t)(void*)(drow + 4 * t);
        asm volatile("global_store_async_from_lds_b128 %0, %1, off th:TH_STORE_NT"
                     :: "v"(gaddr), "v"(lds_addr) : "memory");
      }
    }
    // drain before LDS slot could be reused (kernel end also waits)
    asm volatile("s_wait_asynccnt 0" ::: "memory");
  }
}

// ---------------------------------------------------------------------------
// Launcher. Inputs: [0] feeds fp32 [N*D], [1] repeats int32 [N],
// [2] output_feeds_size (device scalar, unused host-side; rows = out_size/D).
// d_ws holds cum[] (N * 4 bytes = 256 KB).
// ---------------------------------------------------------------------------
extern "C" void kernel_launch(void* const* d_in, const int* in_sizes, int n_in,
                              void* d_out, int out_size, void* d_ws, size_t ws_size,
                              hipStream_t stream) {
  const float* feeds = (const float*)d_in[0];
  const int* repeats = (const int*)d_in[1];
  const int n_src = in_sizes[1];            // 65536
  const int d = in_sizes[0] / n_src;        // 512
  const int out_rows = out_size / d;        // 262144
  int* cum = (int*)d_ws;

  scan_kernel<<<1, SCAN_BLOCK, 0, stream>>>(repeats, cum, n_src);

  const int blocks = (out_rows + ROWS_PER_BLOCK - 1) / ROWS_PER_BLOCK;
  gather_rows_kernel<<<blocks, GATHER_THREADS, 0, stream>>>(
      feeds, cum, (float*)d_out, n_src, d, out_rows);
}